// Fattn_1838246002846
// MI455X (gfx1250) — compile-verified
//
#include <hip/hip_runtime.h>
#include <hip/hip_bf16.h>

typedef float v2f __attribute__((ext_vector_type(2)));
typedef float v8f __attribute__((ext_vector_type(8)));

#define BB   16      // batch
#define CC   256     // channels (== NCF)
#define HWSZ 16384   // 128*128
#define AANUM 64     // attr features
#define TILE 32      // pixels per workgroup

__launch_bounds__(256)
__global__ void fattn_fused(const float* __restrict__ x,
                            const float* __restrict__ attr,
                            const float* __restrict__ wemb,
                            float* __restrict__ out) {
    __shared__ __align__(16) float lds_w[CC];
    __shared__ __align__(16) float lds_attr[BB * AANUM];
    __shared__ __align__(16) float s_lds[BB][TILE];
    __shared__ __align__(16) float q_lds[BB][TILE];

    const int tid  = threadIdx.x;
    const int wave = tid >> 5;
    const int lane = tid & 31;
    const int n0   = blockIdx.x * TILE;

    // ---- Stage small operands into LDS, zero q accumulator ----
    lds_w[tid] = wemb[tid];
    ((float4*)lds_attr)[tid] = ((const float4*)attr)[tid];   // 1024 floats / 256 thr
    ((float*)q_lds)[tid] = 0.0f;
    ((float*)q_lds)[tid + 256] = 0.0f;
    __syncthreads();

    // ---- Phase 1: s[b][n] = sum_c w[c] * x[b,c,n]  via V_WMMA_F32_16X16X4_F32 ----
    // A (16x4 f32): lanes 0-15 hold K=0,1 ; lanes 16-31 hold K=2,3 (M replicated = w_emb)
    // B (4x16 f32): lanes = N (pixel), half-wave/VGPR = K row (x data)
    // D row 0 (== s for 16 pixels) lives in acc[0] of lanes 0-15.
    const int kh = lane >> 4;     // half-wave K selector
    const int nl = lane & 15;     // pixel within 16-wide subtile
    for (int task = wave; task < 32; task += 8) {
        const int b  = task & 15;
        const int st = task >> 4; // which 16-pixel subtile of the 32-pixel tile
        const float* xp = x + (size_t)b * CC * HWSZ
                            + (size_t)(2 * kh) * HWSZ
                            + (size_t)(n0 + st * 16 + nl);
        v8f acc = {0.f, 0.f, 0.f, 0.f, 0.f, 0.f, 0.f, 0.f};
#pragma unroll 4
        for (int kb = 0; kb < CC; kb += 4) {
            v2f af, bf;
            af[0] = lds_w[kb + 2 * kh];
            af[1] = lds_w[kb + 2 * kh + 1];
            bf[0] = xp[0];
            bf[1] = xp[HWSZ];
            acc = __builtin_amdgcn_wmma_f32_16x16x4_f32(
                false, af, false, bf, (short)0, acc, false, false);
            xp += (size_t)4 * HWSZ;
        }
        if (lane < 16) s_lds[b][st * 16 + nl] = acc[0];
    }
    __syncthreads();

    // ---- Phase 2: softmax over batch dim (16) per (a, n); q[b,n] = sum_a attr[b,a]*p ----
    {
        const int n  = tid & (TILE - 1); // pixel in tile
        const int ag = tid >> 5;         // a-group: 8 groups x 8 a-values
        float sv[BB];
#pragma unroll
        for (int b = 0; b < BB; ++b) sv[b] = s_lds[b][n];
        float qloc[BB];
#pragma unroll
        for (int b = 0; b < BB; ++b) qloc[b] = 0.0f;

        for (int aa = 0; aa < 8; ++aa) {
            const int a = ag * 8 + aa;
            float tv[BB];
            float m = -3.402823466e38f;
#pragma unroll
            for (int b = 0; b < BB; ++b) {
                tv[b] = lds_attr[b * AANUM + a] * sv[b];
                m = fmaxf(m, tv[b]);
            }
            float z = 0.0f;
#pragma unroll
            for (int b = 0; b < BB; ++b) {
                tv[b] = __expf(tv[b] - m);
                z += tv[b];
            }
            const float rz = 1.0f / z;
#pragma unroll
            for (int b = 0; b < BB; ++b)
                qloc[b] += lds_attr[b * AANUM + a] * tv[b] * rz;
        }
#pragma unroll
        for (int b = 0; b < BB; ++b) atomicAdd(&q_lds[b][n], qloc[b]);
    }
    __syncthreads();

    // ---- Phase 3: out[b,c,n] = w[c] * q[b,n]  (b128 coalesced stores, HBM-rate) ----
    {
        const int sub = lane >> 3;        // 4 rows per wave-iteration
        const int px4 = (lane & 7) * 4;   // 8 lanes x float4 = 32 pixels per row
        for (int bb2 = 0; bb2 < 2; ++bb2) {
            const int b = wave * 2 + bb2;
            const float4 q4 = *(const float4*)&q_lds[b][px4];
            float* obase = out + (size_t)b * CC * HWSZ + (size_t)(n0 + px4);
#pragma unroll 4
            for (int ci = 0; ci < 64; ++ci) {
                const int c = ci * 4 + sub;
                const float wv = lds_w[c];
                float4 v;
                v.x = wv * q4.x; v.y = wv * q4.y;
                v.z = wv * q4.z; v.w = wv * q4.w;
                *(float4*)(obase + (size_t)c * HWSZ) = v;
            }
        }
    }
}

extern "C" void kernel_launch(void* const* d_in, const int* in_sizes, int n_in,
                              void* d_out, int out_size, void* d_ws, size_t ws_size,
                              hipStream_t stream) {
    (void)in_sizes; (void)n_in; (void)out_size; (void)d_ws; (void)ws_size;
    const float* x    = (const float*)d_in[0];   // (16, 256, 128, 128)
    const float* attr = (const float*)d_in[1];   // (16, 64)
    const float* wemb = (const float*)d_in[2];   // (256,)
    float* out = (float*)d_out;                  // (16, 256, 128, 128)

    dim3 grid(HWSZ / TILE);   // 512 workgroups
    dim3 block(256);          // 8 wave32s
    fattn_fused<<<grid, block, 0, stream>>>(x, attr, wemb, out);
}